// MessagePassingConvolutionHaiku_70583492542673
// MI455X (gfx1250) — compile-verified
//
#include <hip/hip_runtime.h>
#include <hip/hip_bf16.h>

// ---------------------------------------------------------------------------
// e3nn message-passing convolution, fused for MI455X (gfx1250, wave32).
//   node_up    : s_up = ns@W_up_s/8        [N,64]
//                v_up = nv einsum W_up_v / sqrt(32)  [N,32,3]
//   edge_kernel: radial basis -> MLP(8->64 gelu ->192) via V_WMMA_F32_16X16X4_F32
//                gate tensor-product messages, scatter with f32 L2 atomics
//   node_down  : agg@W_down / sqrt(96*8) -> out [N,160]
// ---------------------------------------------------------------------------

#define N_NODES   50000
#define N_EDGES   400000
#define C0        64
#define C1        32
#define MLP_HID   64
#define EPB       128           // edges per block
#define BTHREADS  256           // 8 wave32s -> 8 x 16 edges

typedef __attribute__((ext_vector_type(2))) float v2f;
typedef __attribute__((ext_vector_type(8))) float v8f;

__device__ __forceinline__ void atomAddF(float* p, float v) {
    // relaxed, device-scope fp32 atomic -> global_atomic_add_f32 (no CAS loop)
    __hip_atomic_fetch_add(p, v, __ATOMIC_RELAXED, __HIP_MEMORY_SCOPE_AGENT);
}

__device__ __forceinline__ float gelu_norm(float z) {
    // jax.nn.gelu (tanh approx), normalized to unit 2nd moment (1/GELU_C)
    float t = tanhf(0.7978845608028654f * (z + 0.044715f * z * z * z));
    return 0.5f * z * (1.0f + t) * 1.70151f;
}

// ---------------------------- node linear-up -------------------------------
__global__ void node_up_kernel(const float* __restrict__ ns,
                               const float* __restrict__ nv,
                               const float* __restrict__ Ws,
                               const float* __restrict__ Wv,
                               float* __restrict__ s_up,
                               float* __restrict__ v_up) {
    int i = blockIdx.x * blockDim.x + threadIdx.x;
    const int total_s = N_NODES * C0;
    const int total_v = N_NODES * C1 * 3;
    if (i < total_s) {
        int n = i / C0, m = i % C0;
        const float* row = ns + n * C0;
        float acc = 0.f;
        #pragma unroll 8
        for (int c = 0; c < C0; ++c) acc = fmaf(row[c], Ws[c * C0 + m], acc);
        s_up[i] = acc * 0.125f;                       // 1/sqrt(64)
    } else if (i < total_s + total_v) {
        int j = i - total_s;
        int n = j / (C1 * 3);
        int rm = j % (C1 * 3);
        int m = rm / 3, d = rm % 3;
        const float* row = nv + n * C1 * 3;
        float acc = 0.f;
        #pragma unroll 8
        for (int c = 0; c < C1; ++c) acc = fmaf(row[c * 3 + d], Wv[c * C1 + m], acc);
        v_up[(n * C1 + m) * 3 + d] = acc * 0.17677669529663687f;  // 1/sqrt(32)
    }
}

// ------------------------- fused edge kernel -------------------------------
__global__ void __launch_bounds__(BTHREADS)
edge_kernel(const float* __restrict__ pos,
            const int*   __restrict__ senders,
            const int*   __restrict__ receivers,
            const float* __restrict__ W1,      // [8 x 64]
            const float* __restrict__ W2,      // [64 x 192]
            const float* __restrict__ s_up,    // [N,64]
            const float* __restrict__ v_up,    // [N,32,3]
            float* __restrict__ agg_s,         // [N,96]
            float* __restrict__ agg_v) {       // [N,96,3]
    __shared__ float sh_lds[EPB][3];
    __shared__ float rb_lds[EPB][8];
    __shared__ float h_lds[EPB][MLP_HID];
    __shared__ int   snd_lds[EPB];
    __shared__ int   rcv_lds[EPB];

    const int tid = threadIdx.x;
    const int e0  = blockIdx.x * EPB;

    // Phase 1: geometry + radial basis (one edge per thread 0..127)
    if (tid < EPB) {
        int e = e0 + tid;
        int s = senders[e], r = receivers[e];
        snd_lds[tid] = s;  rcv_lds[tid] = r;
        float dx = pos[r * 3 + 0] - pos[s * 3 + 0];
        float dy = pos[r * 3 + 1] - pos[s * 3 + 1];
        float dz = pos[r * 3 + 2] - pos[s * 3 + 2];
        float rr  = sqrtf(dx * dx + dy * dy + dz * dz);
        float inv = 1.0f / fmaxf(rr, 1e-9f);
        sh_lds[tid][0] = 1.7320508075688772f * dx * inv;   // sqrt(3)*unit
        sh_lds[tid][1] = 1.7320508075688772f * dy * inv;
        sh_lds[tid][2] = 1.7320508075688772f * dz * inv;
        float x  = rr * (1.0f / 6.0f);                     // r / cutoff
        float tt = 2.0f * (1.0f - x);
        float env = (tt > 0.0f) ? 1.9784655268974388f * __expf(-1.0f / tt) : 0.0f;
        #pragma unroll
        for (int b = 0; b < 8; ++b) {
            float np  = (float)(b + 1) * 3.14159265358979f;
            float sn  = (x > 1e-12f) ? __sinf(np * x) / x : np;  // n*pi*sinc(n*x)*pi... limit n*pi
            rb_lds[tid][b] = 1.4142135623730951f * sn * env;
        }
    }
    __syncthreads();

    // Phase 2: MLP hidden layer h = gelu(rb @ W1 / sqrt(8)) / GELU_C  -> LDS
    for (int i = tid; i < EPB * MLP_HID; i += BTHREADS) {
        int e = i >> 6, j = i & 63;
        float acc = 0.f;
        #pragma unroll
        for (int b = 0; b < 8; ++b) acc = fmaf(rb_lds[e][b], W1[b * 64 + j], acc);
        h_lds[e][j] = gelu_norm(acc * 0.3535533905932738f);    // 1/sqrt(8)
    }
    __syncthreads();

    // Phase 3: mix = h @ W2 via WMMA f32 16x16x4; gate + scatter in-register.
    // Wave handles 16 edges; D fragment: lane -> out channel, VGPR r -> edge row.
    const int lane  = tid & 31;
    const int wave  = tid >> 5;
    const int hi    = lane >> 4;         // K/M half select
    const int ln    = lane & 15;
    const int wbase = wave * 16;         // first edge (in block) of this wave
    const int eA    = wbase + ln;        // A-fragment edge (row M = ln)

    #pragma unroll 1
    for (int t = 0; t < 12; ++t) {       // 12 tiles of 16 output channels
        v8f acc = {};
        #pragma unroll
        for (int kk = 0; kk < 16; ++kk) {            // K = 64 in steps of 4
            const int k0 = 4 * kk + 2 * hi;
            v2f a, b;
            a.x = h_lds[eA][k0];
            a.y = h_lds[eA][k0 + 1];
            const int n = t * 16 + ln;
            b.x = W2[(k0    ) * 192 + n];
            b.y = W2[(k0 + 1) * 192 + n];
            acc = __builtin_amdgcn_wmma_f32_16x16x4_f32(
                false, a, false, b, (short)0, acc, false, false);
        }
        const int c = t * 16 + ln;       // global mix channel; range uniform per t
        #pragma unroll
        for (int r = 0; r < 8; ++r) {
            const int   eL = wbase + r + 8 * hi;     // edge row M = r + 8*hi
            const int   s  = snd_lds[eL];
            const int   rc = rcv_lds[eL];
            const float g  = acc[r] * 0.125f;        // fold 1/sqrt(64) of MLP2
            if (t < 4) {                              // c in [0,64): ms gate
                float m = s_up[s * 64 + c];
                atomAddF(&agg_s[rc * 96 + c], m * g);
            } else if (t < 6) {                       // c in [64,96): tp_s gate
                int cc = c - 64;
                const float* vp = &v_up[(s * 32 + cc) * 3];
                float m = (vp[0] * sh_lds[eL][0] + vp[1] * sh_lds[eL][1] +
                           vp[2] * sh_lds[eL][2]) * 0.5773502691896258f;
                atomAddF(&agg_s[rc * 96 + c], m * g);
            } else if (t < 8) {                       // c in [96,128): mv gate
                int cc = c - 96;
                const float* vp = &v_up[(s * 32 + cc) * 3];
                #pragma unroll
                for (int d = 0; d < 3; ++d)
                    atomAddF(&agg_v[(rc * 96 + cc) * 3 + d], vp[d] * g);
            } else {                                  // c in [128,192): tp_v gate
                int cc = c - 128;
                float m = s_up[s * 64 + cc];
                #pragma unroll
                for (int d = 0; d < 3; ++d)
                    atomAddF(&agg_v[(rc * 96 + (cc + 32)) * 3 + d],
                             m * sh_lds[eL][d] * g);
            }
        }
    }
}

// ---------------------------- node linear-down -----------------------------
__global__ void node_down_kernel(const float* __restrict__ agg_s,
                                 const float* __restrict__ agg_v,
                                 const float* __restrict__ Wds,   // [96 x 64]
                                 const float* __restrict__ Wdv,   // [96 x 32]
                                 float* __restrict__ out) {       // [N,160]
    int i = blockIdx.x * blockDim.x + threadIdx.x;
    if (i >= N_NODES * 160) return;
    int n = i / 160, c = i % 160;
    const float scale = 0.03608439182435161f;    // 1/sqrt(96*8)
    float acc = 0.f;
    if (c < 64) {
        const float* row = agg_s + n * 96;
        #pragma unroll 8
        for (int k = 0; k < 96; ++k) acc = fmaf(row[k], Wds[k * 64 + c], acc);
    } else {
        int m = (c - 64) / 3, d = (c - 64) % 3;
        const float* row = agg_v + (size_t)n * 96 * 3 + d;
        #pragma unroll 8
        for (int k = 0; k < 96; ++k) acc = fmaf(row[k * 3], Wdv[k * 32 + m], acc);
    }
    out[i] = acc * scale;
}

// ------------------------------- launcher ----------------------------------
extern "C" void kernel_launch(void* const* d_in, const int* in_sizes, int n_in,
                              void* d_out, int out_size, void* d_ws, size_t ws_size,
                              hipStream_t stream) {
    const float* positions    = (const float*)d_in[0];
    const float* node_scalars = (const float*)d_in[1];
    const float* node_vectors = (const float*)d_in[2];
    const int*   senders      = (const int*)  d_in[3];
    const int*   receivers    = (const int*)  d_in[4];
    const float* W_up_s       = (const float*)d_in[5];
    const float* W_up_v       = (const float*)d_in[6];
    const float* W_mlp1       = (const float*)d_in[7];
    const float* W_mlp2       = (const float*)d_in[8];
    const float* W_down_s     = (const float*)d_in[9];
    const float* W_down_v     = (const float*)d_in[10];
    float* out = (float*)d_out;

    float* ws    = (float*)d_ws;
    float* s_up  = ws;                                  // N*64
    float* v_up  = s_up + (size_t)N_NODES * 64;         // N*96
    float* agg_s = v_up + (size_t)N_NODES * 96;         // N*96
    float* agg_v = agg_s + (size_t)N_NODES * 96;        // N*288

    hipMemsetAsync(agg_s, 0, (size_t)N_NODES * (96 + 288) * sizeof(float), stream);

    const int up_total = N_NODES * (64 + 96);
    node_up_kernel<<<(up_total + 255) / 256, 256, 0, stream>>>(
        node_scalars, node_vectors, W_up_s, W_up_v, s_up, v_up);

    edge_kernel<<<N_EDGES / EPB, BTHREADS, 0, stream>>>(
        positions, senders, receivers, W_mlp1, W_mlp2, s_up, v_up, agg_s, agg_v);

    node_down_kernel<<<(N_NODES * 160 + 255) / 256, 256, 0, stream>>>(
        agg_s, agg_v, W_down_s, W_down_v, out);
}